// FixedAdaptiveSVDPlaneProjection_87033217286475
// MI455X (gfx1250) — compile-verified
//
#include <hip/hip_runtime.h>
#include <hip/hip_bf16.h>
#include <math.h>
#include <stdint.h>

#define NPTS   131072
#define NPL    128
#define NBATCH 4
#define THR    0.01f

typedef float v2f __attribute__((ext_vector_type(2)));
typedef float v8f __attribute__((ext_vector_type(8)));

// ---- workspace layout (floats) ----
// [0, 2048)      prepped planes: [batch][128][4] = unit nx,ny,nz, off' (1e30 if invalid)
// [2048, 7168)   accumulators:   [batch][128][10] = cnt, sx,sy,sz, sxx,sxy,sxz,syy,syz,szz
// [7168, 9216)   refined planes: [batch][128][4] = rn.xyz, rd   (0 if cnt<3)
#define WS_PREP 0
#define WS_ACC  2048
#define WS_REF  7168

// ---- gfx1250 async global->LDS copy (16B), tracked by ASYNCcnt ----------
__device__ __forceinline__ void lds_async_b128(void* lds, const void* gaddr) {
    uint32_t l = (uint32_t)(uintptr_t)lds;   // low 32 bits of flat LDS addr = LDS byte offset
    asm volatile("global_load_async_to_lds_b128 %0, %1, off"
                 :: "v"(l), "v"(gaddr) : "memory");
}
__device__ __forceinline__ void wait_async0() {
    asm volatile("s_wait_asynccnt 0x0" ::: "memory");
}

// -------------------------------------------------------------------------
// Kernel 1: normalize planes + zero accumulators
// -------------------------------------------------------------------------
__global__ void prep_kernel(const float* __restrict__ planes, float* __restrict__ ws) {
    int idx = blockIdx.x * blockDim.x + threadIdx.x;
    int nthreads = gridDim.x * blockDim.x;
    for (int i = idx; i < NBATCH * NPL * 10; i += nthreads) ws[WS_ACC + i] = 0.0f;
    if (idx < NBATCH * NPL) {
        const float* pl = planes + idx * 4;
        float nx = pl[0], ny = pl[1], nz = pl[2], off = pl[3];
        float nrm = sqrtf(nx * nx + ny * ny + nz * nz);
        bool  valid = nrm > 1e-6f;
        float inv = 1.0f / fmaxf(nrm, 1e-12f);
        float* o = ws + WS_PREP + idx * 4;
        o[0] = nx * inv;
        o[1] = ny * inv;
        o[2] = nz * inv;
        o[3] = valid ? off : 1e30f;   // invalid plane -> mask never fires
    }
}

// -------------------------------------------------------------------------
// Kernel 2: per-plane statistics via V_WMMA_F32_16X16X4_F32
//   dist[M][N] = p_M . un_N + off_N  computed as (x,y,z,1) @ (nx,ny,nz,off)
// grid: 2048 blocks x 256 threads; 512 blocks per batch (256 points each)
// -------------------------------------------------------------------------
__global__ void stats_kernel(const float* __restrict__ pts, float* __restrict__ ws) {
    __shared__ float  lp[256 * 3];   // points, AoS (3072 B), staged async
    __shared__ float4 lpl[NPL];      // prepped planes (2048 B), staged async
    __shared__ float  acc[NPL * 10]; // per-plane moment accumulators

    const int tid   = threadIdx.x;
    const int batch = blockIdx.x >> 9;
    const int pbase = (blockIdx.x & 511) << 8;

    // async-stage 256 points (3072 B = 192 x b128) and 128 planes (2048 B = 128 x b128)
    const float* gp = pts + ((size_t)batch * NPTS + pbase) * 3;
    const float* pp = ws + WS_PREP + batch * NPL * 4;
    if (tid < 192) lds_async_b128(&lp[tid * 4], gp + tid * 4);
    if (tid < 128) lds_async_b128(&lpl[tid],    pp + tid * 4);

    for (int i = tid; i < NPL * 10; i += 256) acc[i] = 0.0f;

    wait_async0();        // my wave's async copies landed in LDS
    __syncthreads();      // all waves' copies + acc zeroing visible

    const int  lane  = tid & 31;
    const int  wbase = tid & ~31;        // wave's first point (block-local)
    const int  m     = lane & 15;
    const bool hiK   = lane >= 16;       // lanes 16-31 carry K=2,3 (A) / rows 2,3 (B)

    // B operands: plane group g covers planes g*16 .. g*16+15
    v2f b[8];
#pragma unroll
    for (int g = 0; g < 8; ++g) {
        float4 q = lpl[g * 16 + m];
        v2f bb;
        bb.x = hiK ? q.z : q.x;          // K=0:nx | K=2:nz
        bb.y = hiK ? q.w : q.y;          // K=1:ny | K=3:off
        b[g] = bb;
    }

#pragma unroll
    for (int t = 0; t < 2; ++t) {
        const int tb = wbase + t * 16;   // tile of 16 points
        const int p0 = (tb + m) * 3;
        v2f a;
        a.x = hiK ? lp[p0 + 2] : lp[p0 + 0];   // K=0:x | K=2:z
        a.y = hiK ? 1.0f       : lp[p0 + 1];   // K=1:y | K=3:1
#pragma unroll
        for (int g = 0; g < 8; ++g) {
            v8f c = {0.f, 0.f, 0.f, 0.f, 0.f, 0.f, 0.f, 0.f};
            c = __builtin_amdgcn_wmma_f32_16x16x4_f32(
                    false, a, false, b[g], (short)0, c, false, false);
            const int plane = g * 16 + m;      // this lane's N column
#pragma unroll
            for (int r = 0; r < 8; ++r) {
                float d = c[r];                // point M = r + 8*hiK, plane N
                if (fabsf(d) < THR) {
                    int   pi = (tb + r + (hiK ? 8 : 0)) * 3;
                    float px = lp[pi + 0], py = lp[pi + 1], pz = lp[pi + 2];
                    float* A = acc + plane * 10;
                    atomicAdd(A + 0, 1.0f);
                    atomicAdd(A + 1, px);
                    atomicAdd(A + 2, py);
                    atomicAdd(A + 3, pz);
                    atomicAdd(A + 4, px * px);
                    atomicAdd(A + 5, px * py);
                    atomicAdd(A + 6, px * pz);
                    atomicAdd(A + 7, py * py);
                    atomicAdd(A + 8, py * pz);
                    atomicAdd(A + 9, pz * pz);
                }
            }
        }
    }
    __syncthreads();

    float* gacc = ws + WS_ACC + batch * NPL * 10;
    for (int i = tid; i < NPL * 10; i += 256) {
        float v = acc[i];
        if (v != 0.0f) atomicAdd(gacc + i, v);
    }
}

// -------------------------------------------------------------------------
// Kernel 3: covariance -> smallest eigenvector -> refined plane
// 512 threads, one per (batch, plane)
// -------------------------------------------------------------------------
__global__ void solve_kernel(float* __restrict__ ws) {
    int idx = blockIdx.x * blockDim.x + threadIdx.x;
    if (idx >= NBATCH * NPL) return;

    const float* A = ws + WS_ACC + idx * 10;
    float cnt = A[0];
    float sx = A[1], sy = A[2], sz = A[3];
    float sxx = A[4], sxy = A[5], sxz = A[6], syy = A[7], syz = A[8], szz = A[9];

    float icnt = 1.0f / fmaxf(cnt, 1.0f);
    float cx = sx * icnt, cy = sy * icnt, cz = sz * icnt;
    float idn = 1.0f / fmaxf(cnt - 1.0f, 1.0f);

    // cov = (Spp - c*Sp^T - Sp*c^T + cnt*c*c^T)/max(cnt-1,1) + 1e-6 I
    float a00 = (sxx - 2.0f * cx * sx + cnt * cx * cx) * idn + 1e-6f;
    float a11 = (syy - 2.0f * cy * sy + cnt * cy * cy) * idn + 1e-6f;
    float a22 = (szz - 2.0f * cz * sz + cnt * cz * cz) * idn + 1e-6f;
    float a01 = (sxy - cx * sy - cy * sx + cnt * cx * cy) * idn;
    float a02 = (sxz - cx * sz - cz * sx + cnt * cx * cz) * idn;
    float a12 = (syz - cy * sz - cz * sy + cnt * cy * cz) * idn;

    // smallest eigenvalue (trigonometric method, symmetric 3x3)
    float q3 = (a00 + a11 + a22) * (1.0f / 3.0f);
    float p1 = a01 * a01 + a02 * a02 + a12 * a12;
    float b00 = a00 - q3, b11 = a11 - q3, b22 = a22 - q3;
    float p2 = b00 * b00 + b11 * b11 + b22 * b22 + 2.0f * p1;
    float pd = sqrtf(p2 * (1.0f / 6.0f));
    float lmin;
    if (pd > 1e-20f) {
        float ip  = 1.0f / pd;
        float c00 = b00 * ip, c11 = b11 * ip, c22 = b22 * ip;
        float c01 = a01 * ip, c02 = a02 * ip, c12 = a12 * ip;
        float det = c00 * (c11 * c22 - c12 * c12)
                  - c01 * (c01 * c22 - c12 * c02)
                  + c02 * (c01 * c12 - c11 * c02);
        float rr  = fminf(1.0f, fmaxf(-1.0f, det * 0.5f));
        float phi = acosf(rr) * (1.0f / 3.0f);
        lmin = q3 + 2.0f * pd * cosf(phi + 2.0943951023931953f); // + 2*pi/3
    } else {
        lmin = q3;
    }

    // eigenvector: largest cross product among rows of (A - lmin*I)
    float r0x = a00 - lmin, r0y = a01,        r0z = a02;
    float r1x = a01,        r1y = a11 - lmin, r1z = a12;
    float r2x = a02,        r2y = a12,        r2z = a22 - lmin;

    float v0x = r0y * r1z - r0z * r1y, v0y = r0z * r1x - r0x * r1z, v0z = r0x * r1y - r0y * r1x;
    float v1x = r0y * r2z - r0z * r2y, v1y = r0z * r2x - r0x * r2z, v1z = r0x * r2y - r0y * r2x;
    float v2x = r1y * r2z - r1z * r2y, v2y = r1z * r2x - r1x * r2z, v2z = r1x * r2y - r1y * r2x;
    float n0 = v0x * v0x + v0y * v0y + v0z * v0z;
    float n1 = v1x * v1x + v1y * v1y + v1z * v1z;
    float n2 = v2x * v2x + v2y * v2y + v2z * v2z;

    float vx = v0x, vy = v0y, vz = v0z, best = n0;
    if (n1 > best) { vx = v1x; vy = v1y; vz = v1z; best = n1; }
    if (n2 > best) { vx = v2x; vy = v2y; vz = v2z; best = n2; }

    const float* u = ws + WS_PREP + idx * 4;
    float unx = u[0], uny = u[1], unz = u[2];

    if (best > 1e-30f) {
        float inv = 1.0f / sqrtf(best);
        vx *= inv; vy *= inv; vz *= inv;
    } else {                  // degenerate: fall back to input normal
        vx = unx; vy = uny; vz = unz;
    }
    if (vx * unx + vy * uny + vz * unz < 0.0f) { vx = -vx; vy = -vy; vz = -vz; }
    float rd = -(cx * vx + cy * vy + cz * vz);

    bool active = cnt >= 2.5f;         // cnt >= 3 (cnt is an exact integer in f32)
    float* R = ws + WS_REF + idx * 4;
    R[0] = active ? vx : 0.0f;
    R[1] = active ? vy : 0.0f;
    R[2] = active ? vz : 0.0f;
    R[3] = active ? rd : 0.0f;
}

// -------------------------------------------------------------------------
// Kernel 4: sequential projection, one lane per point, 128 planes in order
// grid: 2048 blocks x 256 threads
// -------------------------------------------------------------------------
__global__ void project_kernel(const float* __restrict__ pts,
                               const float* __restrict__ ws,
                               float* __restrict__ out) {
    __shared__ float4 lu[NPL];   // unit normal + off' (mask test, from ORIGINAL point)
    __shared__ float4 lr[NPL];   // refined normal + rd (0 if inactive)

    const int tid   = threadIdx.x;
    const int batch = blockIdx.x >> 9;
    const int pbase = (blockIdx.x & 511) << 8;

    // async-stage both plane tables: 256 threads x one b128 each
    const float* U = ws + WS_PREP + batch * NPL * 4;
    const float* R = ws + WS_REF  + batch * NPL * 4;
    if (tid < 128) lds_async_b128(&lu[tid], U + tid * 4);
    else           lds_async_b128(&lr[tid - 128], R + (tid - 128) * 4);

    const size_t g  = (size_t)batch * NPTS + pbase + tid;
    const float  ox = pts[g * 3 + 0];
    const float  oy = pts[g * 3 + 1];
    const float  oz = pts[g * 3 + 2];

    wait_async0();
    __syncthreads();

    float px = ox, py = oy, pz = oz;

#pragma unroll 8
    for (int k = 0; k < NPL; ++k) {
        float4 u = lu[k];
        float4 r = lr[k];
        float d0 = fmaf(ox, u.x, fmaf(oy, u.y, fmaf(oz, u.z, u.w)));   // mask distance
        float dp = fmaf(px, r.x, fmaf(py, r.y, fmaf(pz, r.z, r.w)));   // projection dot
        float t  = (fabsf(d0) < THR) ? dp : 0.0f;
        px = fmaf(-r.x, t, px);
        py = fmaf(-r.y, t, py);
        pz = fmaf(-r.z, t, pz);
    }

    float* proj = out;
    float* disp = out + (size_t)NBATCH * NPTS * 3;
    proj[g * 3 + 0] = px;          proj[g * 3 + 1] = py;          proj[g * 3 + 2] = pz;
    disp[g * 3 + 0] = px - ox;     disp[g * 3 + 1] = py - oy;     disp[g * 3 + 2] = pz - oz;
}

// -------------------------------------------------------------------------
extern "C" void kernel_launch(void* const* d_in, const int* in_sizes, int n_in,
                              void* d_out, int out_size, void* d_ws, size_t ws_size,
                              hipStream_t stream) {
    const float* pts    = (const float*)d_in[0];   // (4, 131072, 3) f32
    const float* planes = (const float*)d_in[1];   // (4, 128, 4)    f32
    float*       out    = (float*)d_out;           // proj then disp, each (4,131072,3)
    float*       ws     = (float*)d_ws;            // needs 9216 floats (36 KB)

    (void)in_sizes; (void)n_in; (void)out_size; (void)ws_size;

    prep_kernel   <<<8,    256, 0, stream>>>(planes, ws);
    stats_kernel  <<<2048, 256, 0, stream>>>(pts, ws);
    solve_kernel  <<<2,    256, 0, stream>>>(ws);
    project_kernel<<<2048, 256, 0, stream>>>(pts, ws, out);
}